// Encoder_4733053960478
// MI455X (gfx1250) — compile-verified
//
#include <hip/hip_runtime.h>
#include <hip/hip_bf16.h>

typedef float v2f __attribute__((ext_vector_type(2)));
typedef float v8f __attribute__((ext_vector_type(8)));

#define EPS 1e-5f

// ---------------------------------------------------------------------------
// Zero-fill
// ---------------------------------------------------------------------------
__global__ void fill_zero(float* __restrict__ p, int n) {
    int i = blockIdx.x * blockDim.x + threadIdx.x;
    if (i < n) p[i] = 0.0f;
}

// ---------------------------------------------------------------------------
// Edge scatter-add, d == 1:  agg[dst[e]] += x[src[e]]
// ---------------------------------------------------------------------------
__global__ void scatter_add_d1(const float* __restrict__ X,
                               const int* __restrict__ src,
                               const int* __restrict__ dst,
                               float* __restrict__ AGG,
                               int nEdges) {
    int e = blockIdx.x * blockDim.x + threadIdx.x;
    if (e >= nEdges) return;
    atomicAdd(&AGG[dst[e]], X[src[e]]);
}

// ---------------------------------------------------------------------------
// Edge scatter-add, d multiple of 4: one thread per (edge, 4-feature group).
// Indices loaded once per 4 features; gather is a b128 load; the 4 atomic
// adds hit one contiguous 16B run (same L2 sector).
// ---------------------------------------------------------------------------
__global__ void scatter_add_v4(const float* __restrict__ X,
                               const int* __restrict__ src,
                               const int* __restrict__ dst,
                               float* __restrict__ AGG,
                               int nEdges, int logd) {
    int idx = blockIdx.x * blockDim.x + threadIdx.x;
    const int gshift = logd - 2;                 // groups-per-edge = d/4
    int total = nEdges << gshift;
    if (idx >= total) return;
    int e  = idx >> gshift;
    int f4 = (idx & ((1 << gshift) - 1)) << 2;
    int s = src[e];
    int t = dst[e];
    const float4 v = *(const float4*)(X + ((size_t)s << logd) + f4);
    float* p = AGG + ((size_t)t << logd) + f4;
    atomicAdd(p + 0, v.x);
    atomicAdd(p + 1, v.y);
    atomicAdd(p + 2, v.z);
    atomicAdd(p + 3, v.w);
}

// ---------------------------------------------------------------------------
// GraphConv GEMM with V_WMMA_F32_16X16X4_F32, DIN templated so the K-loop is
// fully unrolled with unconditional float2 (b64) loads and zero branches:
//   OUT[N x dout] = X @ WR^T + AGG @ WN^T + bias
// A-frag (16x4 f32, ISA layout): lanes 0-15 -> K=k+0/k+1, lanes 16-31 -> k+2/k+3.
// B-frag (4x16): B[k][n] = W[n][k] (W is dout x din row-major), same half split.
// C/D (16x16 f32): vgpr r holds rows r (lanes 0-15) and r+8 (lanes 16-31).
// ---------------------------------------------------------------------------
template <int DIN>
__global__ void graphconv_gemm(const float* __restrict__ X,
                               const float* __restrict__ AGG,
                               const float* __restrict__ WR,
                               const float* __restrict__ WN,
                               const float* __restrict__ BIAS,
                               float* __restrict__ OUT,
                               int nRows, int dout) {
    const int lane = threadIdx.x & 31;
    const int wave = threadIdx.x >> 5;
    const int tilesM = nRows >> 4;
    const int tilesN = dout >> 4;
    const int t = blockIdx.x * 8 + wave;          // 8 waves per 256-thread block
    if (t >= tilesM * tilesN) return;             // whole-wave exit: EXEC stays all-1s
    const int tm = t % tilesM;                    // adjacent waves share the same
    const int tn = t / tilesM;                    // weight columns -> L0 hits

    const int half = lane >> 4;                   // 0: lanes 0-15, 1: lanes 16-31
    const int l15  = lane & 15;
    const int rowA = tm * 16 + l15;               // N is a multiple of 16
    const int colB = tn * 16 + l15;               // output channel for this lane

    const float* __restrict__ xrow = X   + (size_t)rowA * DIN;
    const float* __restrict__ grow = AGG + (size_t)rowA * DIN;
    const float* __restrict__ wrw  = WR  + (size_t)colB * DIN;
    const float* __restrict__ wnw  = WN  + (size_t)colB * DIN;

    v8f acc = {};
    if constexpr (DIN == 1) {
        // Single K-step; only K=0 is real. Zero the lanes-16..31 half and the
        // second element with selects (v_cndmask) -- no EXEC manipulation.
        const float xa = xrow[0], ga = grow[0], wa = wrw[0], na = wnw[0];
        v2f a, g, br, bn;
        a.x  = half ? 0.0f : xa;   a.y  = 0.0f;
        g.x  = half ? 0.0f : ga;   g.y  = 0.0f;
        br.x = half ? 0.0f : wa;   br.y = 0.0f;
        bn.x = half ? 0.0f : na;   bn.y = 0.0f;
        acc = __builtin_amdgcn_wmma_f32_16x16x4_f32(false, a, false, br,
                                                    (short)0, acc, false, false);
        acc = __builtin_amdgcn_wmma_f32_16x16x4_f32(false, g, false, bn,
                                                    (short)0, acc, false, false);
    } else {
        static_assert(DIN % 4 == 0, "DIN must be 1 or a multiple of 4");
#pragma unroll
        for (int ks = 0; ks < DIN / 4; ++ks) {
            const int off = ks * 4 + half * 2;    // even, 8B-aligned
            v2f a  = *(const v2f*)(xrow + off);
            v2f g  = *(const v2f*)(grow + off);
            v2f br = *(const v2f*)(wrw + off);
            v2f bn = *(const v2f*)(wnw + off);
            acc = __builtin_amdgcn_wmma_f32_16x16x4_f32(false, a, false, br,
                                                        (short)0, acc, false, false);
            acc = __builtin_amdgcn_wmma_f32_16x16x4_f32(false, g, false, bn,
                                                        (short)0, acc, false, false);
        }
    }

    const float bias = BIAS[colB];
    const int rbase = tm * 16 + half * 8;
#pragma unroll
    for (int r = 0; r < 8; ++r) {
        OUT[(size_t)(rbase + r) * dout + colB] = acc[r] + bias;
    }
}

// ---------------------------------------------------------------------------
// Per-channel sum & sum-of-squares (one block per channel, LDS tree reduce).
// stats[c] = sum, stats[d + c] = sumsq.
// ---------------------------------------------------------------------------
__global__ void col_stats(const float* __restrict__ H,
                          float* __restrict__ stats, int nRows, int d) {
    const int c = blockIdx.x;
    float s = 0.0f, q = 0.0f;
    for (int r = threadIdx.x; r < nRows; r += blockDim.x) {
        float v = H[(size_t)r * d + c];
        s += v;
        q += v * v;
    }
    __shared__ float ss[256];
    __shared__ float qq[256];
    ss[threadIdx.x] = s;
    qq[threadIdx.x] = q;
    __syncthreads();
    for (int off = 128; off > 0; off >>= 1) {
        if ((int)threadIdx.x < off) {
            ss[threadIdx.x] += ss[threadIdx.x + off];
            qq[threadIdx.x] += qq[threadIdx.x + off];
        }
        __syncthreads();
    }
    if (threadIdx.x == 0) {
        stats[c]     = ss[0];
        stats[d + c] = qq[0];
    }
}

// ---------------------------------------------------------------------------
// Fused BatchNorm + GraphNorm + ReLU (in place), all derived from one pass
// of column stats:
//   m = mean(x), v = var(x), s1 = rsqrt(v+eps)*g_bn, b1 = b_bn
//   BN output: mean = b1, var = s1^2 * v
//   GN: m2 = b1, v2 = s1^2*v + b1^2*(1-a)^2
//   out = relu(((x - m)*s1 + b1*(1-a)) * rsqrt(v2+eps) * g_gn + b_gn)
// ---------------------------------------------------------------------------
__global__ void bn_gn_relu(float* __restrict__ H,
                           const float* __restrict__ stats,
                           const float* __restrict__ bng,
                           const float* __restrict__ bnb,
                           const float* __restrict__ gng,
                           const float* __restrict__ gnb,
                           const float* __restrict__ gna,
                           int nRows, int d, int dmask) {
    int idx = blockIdx.x * blockDim.x + threadIdx.x;
    if (idx >= nRows * d) return;
    const int c = idx & dmask;
    const float invN = 1.0f / (float)nRows;
    const float m = stats[c] * invN;
    const float q = stats[d + c] * invN;
    const float v = q - m * m;
    const float s1 = rsqrtf(v + EPS) * bng[c];
    const float b1 = bnb[c];
    const float oma = 1.0f - gna[c];
    const float v2 = s1 * s1 * v + b1 * b1 * oma * oma;
    const float r2 = rsqrtf(v2 + EPS);
    const float x = H[idx];
    float y = ((x - m) * s1 + b1 * oma) * r2 * gng[c] + gnb[c];
    H[idx] = y > 0.0f ? y : 0.0f;
}

// ---------------------------------------------------------------------------
// Host-side orchestration
// ---------------------------------------------------------------------------
extern "C" void kernel_launch(void* const* d_in, const int* in_sizes, int n_in,
                              void* d_out, int out_size, void* d_ws, size_t ws_size,
                              hipStream_t stream) {
    const int N = in_sizes[0];          // x is (N, 1)
    const int E = in_sizes[1] / 2;      // edge_index is (2, E)

    const float* x   = (const float*)d_in[0];
    const int*   ei  = (const int*)d_in[1];
    const int*   src = ei;
    const int*   dst = ei + E;

    const float* c1_wr = (const float*)d_in[2];
    const float* c1_wn = (const float*)d_in[3];
    const float* c1_b  = (const float*)d_in[4];
    const float* c2_wr = (const float*)d_in[5];
    const float* c2_wn = (const float*)d_in[6];
    const float* c2_b  = (const float*)d_in[7];
    const float* c3_wr = (const float*)d_in[8];
    const float* c3_wn = (const float*)d_in[9];
    const float* c3_b  = (const float*)d_in[10];
    const float* cmu_wr = (const float*)d_in[11];
    const float* cmu_wn = (const float*)d_in[12];
    const float* cmu_b  = (const float*)d_in[13];
    const float* clv_wr = (const float*)d_in[14];
    const float* clv_wn = (const float*)d_in[15];
    const float* clv_b  = (const float*)d_in[16];

    const float* bn1_g = (const float*)d_in[17];
    const float* bn1_b = (const float*)d_in[18];
    const float* gn1_g = (const float*)d_in[19];
    const float* gn1_b = (const float*)d_in[20];
    const float* gn1_a = (const float*)d_in[21];
    const float* bn2_g = (const float*)d_in[22];
    const float* bn2_b = (const float*)d_in[23];
    const float* gn2_g = (const float*)d_in[24];
    const float* gn2_b = (const float*)d_in[25];
    const float* gn2_a = (const float*)d_in[26];
    const float* bn3_g = (const float*)d_in[27];
    const float* bn3_b = (const float*)d_in[28];
    const float* gn3_g = (const float*)d_in[29];
    const float* gn3_b = (const float*)d_in[30];
    const float* gn3_a = (const float*)d_in[31];

    float* out_mu = (float*)d_out;                       // N x 64
    float* out_lv = (float*)d_out + (size_t)N * 64;      // N x 64

    // Workspace layout (floats): agg | buf1 | buf2 | stats(128)
    float* ws    = (float*)d_ws;
    float* agg   = ws;
    float* buf1  = ws + (size_t)N * 64;
    float* buf2  = buf1 + (size_t)N * 64;
    float* stats = buf2 + (size_t)N * 64;

    const int TB = 256;
    auto cdiv = [](int a, int b) { return (a + b - 1) / b; };

    // Layer 1: din=1, dout=16
    {
        const int dout = 16;
        fill_zero<<<cdiv(N, TB), TB, 0, stream>>>(agg, N);
        scatter_add_d1<<<cdiv(E, TB), TB, 0, stream>>>(x, src, dst, agg, E);
        int tiles = (N >> 4) * (dout >> 4);
        graphconv_gemm<1><<<cdiv(tiles, 8), TB, 0, stream>>>(x, agg, c1_wr, c1_wn, c1_b,
                                                             buf1, N, dout);
        col_stats<<<dout, TB, 0, stream>>>(buf1, stats, N, dout);
        bn_gn_relu<<<cdiv(N * dout, TB), TB, 0, stream>>>(buf1, stats, bn1_g, bn1_b,
                                                          gn1_g, gn1_b, gn1_a,
                                                          N, dout, dout - 1);
    }
    // Layer 2: din=16, dout=32
    {
        const int din = 16, dout = 32, logd = 4;
        fill_zero<<<cdiv(N * din, TB), TB, 0, stream>>>(agg, N * din);
        scatter_add_v4<<<cdiv(E << (logd - 2), TB), TB, 0, stream>>>(buf1, src, dst,
                                                                     agg, E, logd);
        int tiles = (N >> 4) * (dout >> 4);
        graphconv_gemm<16><<<cdiv(tiles, 8), TB, 0, stream>>>(buf1, agg, c2_wr, c2_wn,
                                                              c2_b, buf2, N, dout);
        col_stats<<<dout, TB, 0, stream>>>(buf2, stats, N, dout);
        bn_gn_relu<<<cdiv(N * dout, TB), TB, 0, stream>>>(buf2, stats, bn2_g, bn2_b,
                                                          gn2_g, gn2_b, gn2_a,
                                                          N, dout, dout - 1);
    }
    // Layer 3: din=32, dout=64
    {
        const int din = 32, dout = 64, logd = 5;
        fill_zero<<<cdiv(N * din, TB), TB, 0, stream>>>(agg, N * din);
        scatter_add_v4<<<cdiv(E << (logd - 2), TB), TB, 0, stream>>>(buf2, src, dst,
                                                                     agg, E, logd);
        int tiles = (N >> 4) * (dout >> 4);
        graphconv_gemm<32><<<cdiv(tiles, 8), TB, 0, stream>>>(buf2, agg, c3_wr, c3_wn,
                                                              c3_b, buf1, N, dout);
        col_stats<<<dout, TB, 0, stream>>>(buf1, stats, N, dout);
        bn_gn_relu<<<cdiv(N * dout, TB), TB, 0, stream>>>(buf1, stats, bn3_g, bn3_b,
                                                          gn3_g, gn3_b, gn3_a,
                                                          N, dout, dout - 1);
    }
    // Heads: mu / logvar share the same segment_sum of h3 (din=64, dout=64)
    {
        const int din = 64, dout = 64, logd = 6;
        fill_zero<<<cdiv(N * din, TB), TB, 0, stream>>>(agg, N * din);
        scatter_add_v4<<<cdiv(E << (logd - 2), TB), TB, 0, stream>>>(buf1, src, dst,
                                                                     agg, E, logd);
        int tiles = (N >> 4) * (dout >> 4);
        graphconv_gemm<64><<<cdiv(tiles, 8), TB, 0, stream>>>(buf1, agg, cmu_wr, cmu_wn,
                                                              cmu_b, out_mu, N, dout);
        graphconv_gemm<64><<<cdiv(tiles, 8), TB, 0, stream>>>(buf1, agg, clv_wr, clv_wn,
                                                              clv_b, out_lv, N, dout);
    }
}